// GCNForDialog_29557964931228
// MI455X (gfx1250) — compile-verified
//
#include <hip/hip_runtime.h>

typedef float v2f __attribute__((ext_vector_type(2)));
typedef float v4f __attribute__((ext_vector_type(4)));
typedef float v8f __attribute__((ext_vector_type(8)));

#define N_NODES 32768
#define E_EDGES 262144
#define F_DIM   128
#define L_LAYERS 64

__device__ __forceinline__ float relu1(float x) {
    // single-instruction ReLU: v_med3_num_f32(x, 0, +MAX) == clamp(x, 0, inf)
    return __builtin_amdgcn_fmed3f(x, 0.0f, 3.402823466e+38f);
}

// ---------------- degree / norm precompute ----------------

__global__ __launch_bounds__(256) void deg_init(float* __restrict__ deg) {
    int i = blockIdx.x * 256 + threadIdx.x;
    if (i < N_NODES) deg[i] = 1.0f;            // self-loop contribution
}

__global__ __launch_bounds__(256) void deg_accum(const long long* __restrict__ dst,
                                                 float* __restrict__ deg) {
    int e = blockIdx.x * 256 + threadIdx.x;
    if (e < E_EDGES) atomicAdd(&deg[(int)dst[e]], 1.0f);
}

__global__ __launch_bounds__(256) void dinv_kernel(const float* __restrict__ deg,
                                                   float* __restrict__ dinv) {
    int i = blockIdx.x * 256 + threadIdx.x;
    if (i < N_NODES) {
        float d = deg[i];
        dinv[i] = (d > 0.0f) ? rsqrtf(d) : 0.0f;
    }
}

__global__ __launch_bounds__(256) void norm_kernel(const long long* __restrict__ src,
                                                   const long long* __restrict__ dst,
                                                   const float* __restrict__ dinv,
                                                   float* __restrict__ nrm) {
    int e = blockIdx.x * 256 + threadIdx.x;
    if (e < E_EDGES) nrm[e] = dinv[(int)src[e]] * dinv[(int)dst[e]];
}

// ---------------- fused dense transform ----------------
// hw = relu_maybe(hin) @ W      via V_WMMA_F32_16X16X4_F32 (exact f32 semantics)
// agg = bias + hw * dinv^2      (bias + self-loop edge fused into the store)
//
// One wave computes a 16-row strip x 128 cols: 8 v8f accumulators.
// A layout (16x4 f32): lanes 0-15 hold {K=kb, kb+1} of row m0+l; lanes 16-31 hold {kb+2, kb+3}.
// B layout (4x16 f32): VGPR0 = row kb (lanes 0-15) / row kb+2 (lanes 16-31); VGPR1 = kb+1 / kb+3.
// D layout (16x16 f32): VGPR r holds row r (lanes 0-15) / row r+8 (lanes 16-31), col = lane&15.

template <bool RELU>
__global__ __launch_bounds__(256) void gemm128_wmma_fused(const float* __restrict__ hin,
                                                          const float* __restrict__ W,
                                                          const float* __restrict__ bias,
                                                          const float* __restrict__ dinv,
                                                          float* __restrict__ hw,
                                                          float* __restrict__ agg) {
    int lane = threadIdx.x & 31;
    int wave = threadIdx.x >> 5;
    int half = lane >> 4;      // 0 or 1
    int l    = lane & 15;
    int m0   = (blockIdx.x * 8 + wave) * 16;

    const float* arow = hin + (size_t)(m0 + l) * F_DIM;
    v8f acc[8] = {};

    for (int kb = 0; kb < F_DIM; kb += 4) {
        v2f a = *(const v2f*)(arow + kb + 2 * half);
        if (RELU) {                     // previous layer's ReLU, fused into the load
            a.x = relu1(a.x);
            a.y = relu1(a.y);
        }
        const float* wr = W + (size_t)(kb + 2 * half) * F_DIM + l;
#pragma unroll
        for (int t = 0; t < 8; ++t) {
            v2f b;
            b.x = wr[t * 16];
            b.y = wr[F_DIM + t * 16];
            acc[t] = __builtin_amdgcn_wmma_f32_16x16x4_f32(
                false, a, false, b, (short)0, acc[t], false, false);
        }
    }

    // per-row self-loop scale dinv^2 for the 8 rows this half-wave owns
    float sl[8];
#pragma unroll
    for (int r = 0; r < 8; ++r) {
        float di = dinv[m0 + half * 8 + r];
        sl[r] = di * di;
    }

#pragma unroll
    for (int t = 0; t < 8; ++t) {
        int    col  = t * 16 + l;
        float  bval = bias[col];
        size_t base = (size_t)(m0 + half * 8) * F_DIM + col;
#pragma unroll
        for (int r = 0; r < 8; ++r) {
            float v = acc[t][r];
            hw[base + (size_t)r * F_DIM]  = v;
            agg[base + (size_t)r * F_DIM] = bval + v * sl[r];   // bias + self-loop init
        }
    }
}

// ---------------- edge scatter: one wave per edge, float4 per lane + 4 atomics ----------------

__global__ __launch_bounds__(256) void scatter_edges(const long long* __restrict__ src,
                                                     const long long* __restrict__ dst,
                                                     const float* __restrict__ nrm,
                                                     const float* __restrict__ hw,
                                                     float* __restrict__ agg) {
    int wave = threadIdx.x >> 5;
    int lane = threadIdx.x & 31;
    int e = blockIdx.x * 8 + wave;
    if (e >= E_EDGES) return;
    int s = (int)src[e];
    int d = (int)dst[e];
    float w = nrm[e];
    v4f v = *(const v4f*)(hw + (size_t)s * F_DIM + lane * 4);
    float* p = agg + (size_t)d * F_DIM + lane * 4;
    atomicAdd(p + 0, v.x * w);
    atomicAdd(p + 1, v.y * w);
    atomicAdd(p + 2, v.z * w);
    atomicAdd(p + 3, v.w * w);
}

// ---------------- residual + classifier: out = (relu(agg) + x) @ cls_w + cls_b ----------------

__global__ __launch_bounds__(256) void classify(const float* __restrict__ agg,
                                                const float* __restrict__ x,
                                                const float* __restrict__ cw,
                                                const float* __restrict__ cb,
                                                float* __restrict__ out) {
    int i = blockIdx.x * 256 + threadIdx.x;
    if (i >= N_NODES) return;
    float o0 = cb[0], o1 = cb[1], o2 = cb[2], o3 = cb[3];
    const float* hr = agg + (size_t)i * F_DIM;
    const float* xr = x + (size_t)i * F_DIM;
#pragma unroll 4
    for (int f = 0; f < F_DIM; ++f) {
        float v = relu1(hr[f]) + xr[f];   // final layer ReLU fused here
        const float* w = cw + f * 4;
        o0 += v * w[0];
        o1 += v * w[1];
        o2 += v * w[2];
        o3 += v * w[3];
    }
    v4f o = {o0, o1, o2, o3};
    *(v4f*)(out + (size_t)i * 4) = o;
}

// ---------------- driver ----------------

extern "C" void kernel_launch(void* const* d_in, const int* in_sizes, int n_in,
                              void* d_out, int out_size, void* d_ws, size_t ws_size,
                              hipStream_t stream) {
    const float*     x       = (const float*)d_in[0];
    const long long* eidx    = (const long long*)d_in[1];   // int64 [2, E]
    const long long* src     = eidx;
    const long long* dst     = eidx + E_EDGES;
    const float*     weights = (const float*)d_in[2];       // [64,128,128]
    const float*     biases  = (const float*)d_in[3];       // [64,128]
    const float*     cls_w   = (const float*)d_in[4];       // [128,4]
    const float*     cls_b   = (const float*)d_in[5];       // [4]
    float*           out     = (float*)d_out;

    char* ws = (char*)d_ws;
    const size_t feat_bytes = (size_t)N_NODES * F_DIM * sizeof(float);  // 16 MB
    float* hw   = (float*)ws;                          // transformed features
    float* hbuf = (float*)(ws + feat_bytes);           // agg (pre-ReLU activations)
    float* deg  = (float*)(ws + 2 * feat_bytes);       // N floats
    float* dinv = deg + N_NODES;                       // N floats
    float* nrm  = dinv + N_NODES;                      // E floats

    // GCN normalization precompute (once)
    deg_init   <<<N_NODES / 256, 256, 0, stream>>>(deg);
    deg_accum  <<<E_EDGES / 256, 256, 0, stream>>>(dst, deg);
    dinv_kernel<<<N_NODES / 256, 256, 0, stream>>>(deg, dinv);
    norm_kernel<<<E_EDGES / 256, 256, 0, stream>>>(src, dst, dinv, nrm);

    for (int l = 0; l < L_LAYERS; ++l) {
        const float* Wl = weights + (size_t)l * F_DIM * F_DIM;
        const float* bl = biases + (size_t)l * F_DIM;
        if (l == 0) {
            gemm128_wmma_fused<false><<<N_NODES / (16 * 8), 256, 0, stream>>>(
                x, Wl, bl, dinv, hw, hbuf);
        } else {
            gemm128_wmma_fused<true><<<N_NODES / (16 * 8), 256, 0, stream>>>(
                hbuf, Wl, bl, dinv, hw, hbuf);
        }
        scatter_edges<<<E_EDGES / 8, 256, 0, stream>>>(src, dst, nrm, hw, hbuf);
    }

    classify<<<N_NODES / 256, 256, 0, stream>>>(hbuf, x, cls_w, cls_b, out);
}